// SelfAttention_22978075034141
// MI455X (gfx1250) — compile-verified
//
#include <hip/hip_runtime.h>

typedef __attribute__((ext_vector_type(2))) float v2f;
typedef __attribute__((ext_vector_type(8))) float v8f;
typedef int v4i_vs __attribute__((vector_size(16)));   // matches builtin's param type

#define D_MODEL 1024
#define NH      16
#define HD      64
#define S_LEN   2048
#define BATCH   2

// ---------------------------------------------------------------------------
// Global -> LDS staging, 16 bytes per lane.
// Preferred path: GLOBAL_LOAD_ASYNC_TO_LDS_B128 (tracked on ASYNCcnt, no VGPR
// round-trip, overlaps with the ds_load/v_wmma compute stream).
// Fallback: synchronous float4 load + ds_store (same double-buffer structure).
// ---------------------------------------------------------------------------
#if __has_builtin(__builtin_amdgcn_global_load_async_to_lds_b128)
#define HAVE_ASYNC_LDS 1
#else
#define HAVE_ASYNC_LDS 0
#endif

__device__ __forceinline__ void stage16(const float* __restrict__ g, float* l) {
#if HAVE_ASYNC_LDS
    __builtin_amdgcn_global_load_async_to_lds_b128(
        (__attribute__((address_space(1))) v4i_vs*)g,
        (__attribute__((address_space(3))) v4i_vs*)l,
        0, 0);
#else
    *(float4*)l = *(const float4*)g;
#endif
}

__device__ __forceinline__ void stage_wait() {
#if HAVE_ASYNC_LDS
    asm volatile("s_wait_asynccnt 0" ::: "memory");
#endif
}

// ---------------------------------------------------------------------------
// Generic fp32 WMMA GEMM:  C[M,N] = A[M,K] @ W[K,N] + bias[N]
//   MODE 0: plain row-major store into out[M*N]
//   MODE 1: QKV scatter store into ws laid out [which][b][h][s][hd],
//           with Q (which==0) pre-scaled by 1/sqrt(HD) = 0.125
// Block: 256 threads = 8 waves. Block tile 64(M) x 64(N), K-step 32,
// double-buffered LDS staging (one barrier per K-step).
// ---------------------------------------------------------------------------
__device__ __forceinline__ void store_qkv(float* __restrict__ ws, int row, int col, float v) {
    int which = col >> 10;          // 0=Q 1=K 2=V
    int dcol  = col & 1023;
    int h     = dcol >> 6;
    int hd    = dcol & 63;
    int b     = row >> 11;          // S_LEN = 2048
    int s     = row & 2047;
    if (which == 0) v *= 0.125f;    // fold 1/sqrt(64) into Q
    size_t idx = ((((size_t)which * BATCH + b) * NH + h) * S_LEN + s) * HD + hd;
    ws[idx] = v;
}

template<int MODE>
__global__ __launch_bounds__(256)
void gemm_f32_wmma(const float* __restrict__ A, const float* __restrict__ W,
                   const float* __restrict__ bias, float* __restrict__ out,
                   int M, int N, int K)
{
    __shared__ float As[2][64][36];   // pad 36: conflict-free frag reads, 16B-aligned rows
    __shared__ float Ws[2][32][68];   // pad 68: float4-aligned rows

    const int t    = threadIdx.x;
    const int lane = t & 31;
    const int wave = t >> 5;
    const int wm   = wave & 3;
    const int wn   = wave >> 2;
    const int m0   = blockIdx.y * 64;
    const int n0   = blockIdx.x * 64;

    const int half = lane >> 4;     // which 16-lane half
    const int l15  = lane & 15;
    const int koff = half * 2;      // K sub-offset per A/B fragment layout

    // staging coordinates
    const int arow = t >> 3, ac4 = (t & 7) * 4;     // A tile 64x32: 2 rounds
    const int wrow = t >> 4, wc4 = (t & 15) * 4;    // W tile 32x64: 2 rounds

    v8f c0 = {}; v8f c1 = {};

    auto stage_tiles = [&](int buf, int k0) {
        #pragma unroll
        for (int i = 0; i < 2; ++i) {
            int r = arow + i * 32;
            stage16(&A[(size_t)(m0 + r) * K + k0 + ac4], &As[buf][r][ac4]);
        }
        #pragma unroll
        for (int i = 0; i < 2; ++i) {
            int r = wrow + i * 16;
            stage16(&W[(size_t)(k0 + r) * N + n0 + wc4], &Ws[buf][r][wc4]);
        }
    };

    // prologue: stage first tiles into buffer 0
    stage_tiles(0, 0);
    stage_wait();
    __syncthreads();

    int buf = 0;
    for (int k0 = 0; k0 < K; k0 += 32) {
        // prefetch next K-step into the other buffer (all waves finished
        // reading it before the barrier that ended the previous step)
        if (k0 + 32 < K) stage_tiles(buf ^ 1, k0 + 32);

        #pragma unroll
        for (int kk = 0; kk < 32; kk += 4) {
            // A fragment: lane holds (m = l15, k = kk+koff .. +1)
            v2f a = *(const v2f*)&As[buf][wm * 16 + l15][kk + koff];
            // B fragments: lane holds (k = kk+koff .. +1, n = l15)
            v2f b0, b1;
            b0.x = Ws[buf][kk + koff][wn * 32 + l15];
            b0.y = Ws[buf][kk + koff + 1][wn * 32 + l15];
            b1.x = Ws[buf][kk + koff][wn * 32 + 16 + l15];
            b1.y = Ws[buf][kk + koff + 1][wn * 32 + 16 + l15];
            c0 = __builtin_amdgcn_wmma_f32_16x16x4_f32(false, a, false, b0, (short)0, c0, false, false);
            c1 = __builtin_amdgcn_wmma_f32_16x16x4_f32(false, a, false, b1, (short)0, c1, false, false);
        }

        stage_wait();        // drain async stores into buf^1
        __syncthreads();     // publish to all waves / protect buf reuse
        buf ^= 1;
    }

    // ---- store: C layout = VGPR r holds row (r + 8*half), col = l15 ----
    #pragma unroll
    for (int r = 0; r < 8; ++r) {
        int row  = m0 + wm * 16 + r + 8 * half;
        int col0 = n0 + wn * 32 + l15;
        int col1 = col0 + 16;
        float v0 = c0[r] + bias[col0];
        float v1 = c1[r] + bias[col1];
        if (MODE == 0) {
            out[(size_t)row * N + col0] = v0;
            out[(size_t)row * N + col1] = v1;
        } else {
            store_qkv(out, row, col0, v0);
            store_qkv(out, row, col1, v1);
        }
    }
}

// ---------------------------------------------------------------------------
// Flash attention, fp32 WMMA. One block per (b*NH+h, 128-query slab).
// 8 waves, each owns 16 queries; K/V 16-row tiles double-buffered in LDS.
// ---------------------------------------------------------------------------
__global__ __launch_bounds__(256)
void attn_f32_wmma(const float* __restrict__ qkv, float* __restrict__ ao)
{
    __shared__ float Ks[2][16][68];       // K tiles, padded (float4-aligned rows)
    __shared__ float Vs[2][16][64];       // V tiles (frag reads are lane-consecutive)
    __shared__ float Ps[8][16][17];       // per-wave P relayout buffer

    const int t     = threadIdx.x;
    const int lane  = t & 31;
    const int wave  = t >> 5;
    const int half  = lane >> 4;
    const int l15   = lane & 15;
    const int dhalf = half * 2;
    const int bh    = blockIdx.y;                    // b*NH + h
    const int qbase = blockIdx.x * 128 + wave * 16;

    const float* Qp = qkv + (size_t)bh * S_LEN * HD;                         // which=0
    const float* Kp = qkv + ((size_t)BATCH * NH + bh) * S_LEN * HD;          // which=1
    const float* Vp = qkv + ((size_t)2 * BATCH * NH + bh) * S_LEN * HD;      // which=2

    // staging coordinates: one float4 of K and one of V per thread
    const int sr = t >> 4, sc4 = (t & 15) * 4;

    auto stage_kv = [&](int buf, int j) {
        stage16(&Kp[((size_t)j * 16 + sr) * HD + sc4], &Ks[buf][sr][sc4]);
        stage16(&Vp[((size_t)j * 16 + sr) * HD + sc4], &Vs[buf][sr][sc4]);
    };

    // Resident Q fragments (already scaled by 1/sqrt(HD)); A-frag layout over d.
    v2f qa[16];
    {
        const float* qrow = Qp + (size_t)(qbase + l15) * HD;
        #pragma unroll
        for (int i = 0; i < 16; ++i)
            qa[i] = *(const v2f*)&qrow[i * 4 + dhalf];
    }

    v8f o0 = {}, o1 = {}, o2 = {}, o3 = {};
    float m_i[8], l_i[8];
    #pragma unroll
    for (int r = 0; r < 8; ++r) { m_i[r] = -3.0e38f; l_i[r] = 0.0f; }

    // prologue: stage tile 0
    stage_kv(0, 0);
    stage_wait();
    __syncthreads();

    int buf = 0;
    const int NJ = S_LEN / 16;
    for (int j = 0; j < NJ; ++j) {
        // prefetch next K/V tile into the other buffer
        if (j + 1 < NJ) stage_kv(buf ^ 1, j + 1);

        // ---- S = Q @ K^T  (16 queries x 16 keys), accumulate over d in steps of 4
        v8f s = {};
        #pragma unroll
        for (int i = 0; i < 16; ++i) {
            v2f b = *(const v2f*)&Ks[buf][l15][i * 4 + dhalf];   // B(k=d, n=key): K[key][d]
            s = __builtin_amdgcn_wmma_f32_16x16x4_f32(false, qa[i], false, b, (short)0, s, false, false);
        }

        // ---- online softmax: row r lives in VGPR r across a 16-lane half
        #pragma unroll
        for (int r = 0; r < 8; ++r) {
            float v = s[r];
            #pragma unroll
            for (int off = 8; off >= 1; off >>= 1)
                v = fmaxf(v, __shfl_xor(v, off, 32));
            float mnew  = fmaxf(m_i[r], v);
            float alpha = __expf(m_i[r] - mnew);
            float p     = __expf(s[r] - mnew);
            float sum   = p;
            #pragma unroll
            for (int off = 8; off >= 1; off >>= 1)
                sum += __shfl_xor(sum, off, 32);
            l_i[r] = l_i[r] * alpha + sum;
            m_i[r] = mnew;
            o0[r] *= alpha; o1[r] *= alpha; o2[r] *= alpha; o3[r] *= alpha;
            Ps[wave][r + 8 * half][l15] = p;     // C-layout -> LDS
        }
        // same-wave LDS RAW: ds ops are in-order, just drain the counter
        asm volatile("s_wait_dscnt 0" ::: "memory");

        // ---- O += P @ V : A = P (relayout from LDS), B = V tile
        #pragma unroll
        for (int kk = 0; kk < 16; kk += 4) {
            v2f a;
            a.x = Ps[wave][l15][kk + dhalf];
            a.y = Ps[wave][l15][kk + dhalf + 1];
            int k0 = kk + dhalf, k1 = k0 + 1;
            v2f b0, b1, b2, b3;
            b0.x = Vs[buf][k0][ 0 + l15];  b0.y = Vs[buf][k1][ 0 + l15];
            b1.x = Vs[buf][k0][16 + l15];  b1.y = Vs[buf][k1][16 + l15];
            b2.x = Vs[buf][k0][32 + l15];  b2.y = Vs[buf][k1][32 + l15];
            b3.x = Vs[buf][k0][48 + l15];  b3.y = Vs[buf][k1][48 + l15];
            o0 = __builtin_amdgcn_wmma_f32_16x16x4_f32(false, a, false, b0, (short)0, o0, false, false);
            o1 = __builtin_amdgcn_wmma_f32_16x16x4_f32(false, a, false, b1, (short)0, o1, false, false);
            o2 = __builtin_amdgcn_wmma_f32_16x16x4_f32(false, a, false, b2, (short)0, o2, false, false);
            o3 = __builtin_amdgcn_wmma_f32_16x16x4_f32(false, a, false, b3, (short)0, o3, false, false);
        }

        stage_wait();        // drain async stores into buf^1
        __syncthreads();     // publish to all waves / protect buf reuse
        buf ^= 1;
    }

    // ---- finalize: O /= l, write back to [B, S, D_MODEL] for the out-projection
    int b = bh >> 4;      // NH == 16
    int h = bh & 15;
    #pragma unroll
    for (int r = 0; r < 8; ++r) {
        float inv = 1.0f / l_i[r];
        int qrow = qbase + r + 8 * half;
        float* orow = ao + ((size_t)b * S_LEN + qrow) * D_MODEL + h * HD;
        orow[ 0 + l15] = o0[r] * inv;
        orow[16 + l15] = o1[r] * inv;
        orow[32 + l15] = o2[r] * inv;
        orow[48 + l15] = o3[r] * inv;
    }
}

// ---------------------------------------------------------------------------
extern "C" void kernel_launch(void* const* d_in, const int* in_sizes, int n_in,
                              void* d_out, int out_size, void* d_ws, size_t ws_size,
                              hipStream_t stream)
{
    (void)in_sizes; (void)n_in; (void)out_size; (void)ws_size;
    const float* x    = (const float*)d_in[0];   // [2, 2048, 1024]
    const float* Wqkv = (const float*)d_in[1];   // [1024, 3072]
    const float* bqkv = (const float*)d_in[2];   // [3072]
    const float* Wout = (const float*)d_in[3];   // [1024, 1024]
    const float* bout = (const float*)d_in[4];   // [1024]
    float* out = (float*)d_out;                  // [2, 2048, 1024]

    float* ws  = (float*)d_ws;
    float* qkv = ws;                                               // 3*2*16*2048*64 = 12,582,912 f32
    float* ao  = ws + (size_t)3 * BATCH * NH * S_LEN * HD;         // 4096*1024     =  4,194,304 f32

    const int M = BATCH * S_LEN;     // 4096
    dim3 blk(256);

    // 1) QKV projection, scatter to [which][b][h][s][hd], Q pre-scaled
    gemm_f32_wmma<1><<<dim3((3 * D_MODEL) / 64, M / 64), blk, 0, stream>>>(
        x, Wqkv, bqkv, qkv, M, 3 * D_MODEL, D_MODEL);

    // 2) flash attention per (b,h), 128 queries per block
    attn_f32_wmma<<<dim3(S_LEN / 128, BATCH * NH), blk, 0, stream>>>(qkv, ao);

    // 3) output projection
    gemm_f32_wmma<0><<<dim3(D_MODEL / 64, M / 64), blk, 0, stream>>>(
        ao, Wout, bout, out, M, D_MODEL, D_MODEL);
}